// RelationSpecificGATConv_90202903151258
// MI455X (gfx1250) — compile-verified
//
#include <hip/hip_runtime.h>

typedef __attribute__((ext_vector_type(2))) float v2f;
typedef __attribute__((ext_vector_type(8))) float v8f;

#define IN_DIM   128
#define OUT_DIM  64
#define HEADS    4
#define HC       256          // HEADS*OUT_DIM
#define NEG_SLOPE 0.2f
#define SM_EPS    1e-16f

__device__ __forceinline__ float leaky(float v) {
    return v > 0.f ? v : NEG_SLOPE * v;
}

// monotone float atomic-max via signed/unsigned int atomics (init must be -inf)
__device__ __forceinline__ void atomicMaxF32(float* addr, float val) {
    if (val >= 0.f) atomicMax((int*)addr, __float_as_int(val));
    else            atomicMin((unsigned int*)addr, __float_as_uint(val));
}

// ---------------------------------------------------------------------------
// 0) init: out = bias (broadcast), amax = -inf, denom = 0
// ---------------------------------------------------------------------------
__global__ void gat_init(float* __restrict__ out, float* __restrict__ amax,
                         float* __restrict__ denom, const float* __restrict__ bias,
                         int ndst) {
    int i = blockIdx.x * blockDim.x + threadIdx.x;
    if (i < ndst * OUT_DIM) out[i] = bias[i & (OUT_DIM - 1)];
    if (i < ndst * HEADS) {
        amax[i]  = -__builtin_inff();
        denom[i] = 0.f;
    }
}

// ---------------------------------------------------------------------------
// 1) fold attention vector into projection: weff[k][h] = sum_c W[k][h*64+c]*att[h][c]
// ---------------------------------------------------------------------------
__global__ void gat_weff(const float* __restrict__ W, const float* __restrict__ att,
                         float* __restrict__ weff) {
    int k = threadIdx.x;   // 0..127
#pragma unroll
    for (int h = 0; h < HEADS; ++h) {
        float s = 0.f;
#pragma unroll 8
        for (int c = 0; c < OUT_DIM; ++c)
            s += W[(size_t)k * HC + h * OUT_DIM + c] * att[h * OUT_DIM + c];
        weff[k * HEADS + h] = s;
    }
}

// ---------------------------------------------------------------------------
// 2) hs = x_src @ W_src via V_WMMA_F32_16X16X4_F32.
//    One wave -> 16(M) x 64(N) tile; block = 4 waves = full 16 x 256 M-tile.
// ---------------------------------------------------------------------------
__global__ __launch_bounds__(128) void gat_gemm_hs(const float* __restrict__ X,
                                                   const float* __restrict__ W,
                                                   float* __restrict__ HS, int M) {
    int wave = blockIdx.x * 4 + (threadIdx.x >> 5);
    int lane = threadIdx.x & 31;
    int mt = wave >> 2;          // M tile
    int nt = wave & 3;           // N tile (64 wide)
    int m0 = mt * 16;
    if (m0 >= M) return;         // wave-uniform
    int n0 = nt * 64;
    int half = lane >> 4;        // K sub-select / M-half for C
    int lm   = lane & 15;

    v8f c0 = {}, c1 = {}, c2 = {}, c3 = {};
    const float* xrow = X + (size_t)(m0 + lm) * IN_DIM;

    for (int k0 = 0; k0 < IN_DIM; k0 += 4) {
        int ka = k0 + 2 * half;
        v2f a;                               // A 16x4: lane<16: K=0,1 ; lane>=16: K=2,3
        a.x = xrow[ka];
        a.y = xrow[ka + 1];
        const float* w0 = W + (size_t)ka * HC + n0 + lm;        // row ka
        const float* w1 = w0 + HC;                              // row ka+1
        v2f b0, b1, b2, b3;                  // B 4x16 per 16-col subtile
        b0.x = w0[0];  b0.y = w1[0];
        b1.x = w0[16]; b1.y = w1[16];
        b2.x = w0[32]; b2.y = w1[32];
        b3.x = w0[48]; b3.y = w1[48];
        c0 = __builtin_amdgcn_wmma_f32_16x16x4_f32(false, a, false, b0, (short)0, c0, false, false);
        c1 = __builtin_amdgcn_wmma_f32_16x16x4_f32(false, a, false, b1, (short)0, c1, false, false);
        c2 = __builtin_amdgcn_wmma_f32_16x16x4_f32(false, a, false, b2, (short)0, c2, false, false);
        c3 = __builtin_amdgcn_wmma_f32_16x16x4_f32(false, a, false, b3, (short)0, c3, false, false);
    }

    // C/D layout: VGPR r, lanes 0-15 -> M=r ; lanes 16-31 -> M=r+8 ; N = lane%16
#pragma unroll
    for (int r = 0; r < 8; ++r) {
        float* orow = HS + (size_t)(m0 + r + 8 * half) * HC + n0 + lm;
        orow[0]  = c0[r];
        orow[16] = c1[r];
        orow[32] = c2[r];
        orow[48] = c3[r];
    }
}

// ---------------------------------------------------------------------------
// 3) per-node logits: a[n][h] = x[n,:] . weff[:,h]
// ---------------------------------------------------------------------------
__global__ void gat_node_logits(const float* __restrict__ X, const float* __restrict__ weff,
                                float* __restrict__ a, int N) {
    int n = blockIdx.x * blockDim.x + threadIdx.x;
    if (n >= N) return;
    const float* xr = X + (size_t)n * IN_DIM;
    float s0 = 0.f, s1 = 0.f, s2 = 0.f, s3 = 0.f;
#pragma unroll 8
    for (int k = 0; k < IN_DIM; ++k) {
        float xv = xr[k];
        s0 += xv * weff[k * HEADS + 0];
        s1 += xv * weff[k * HEADS + 1];
        s2 += xv * weff[k * HEADS + 2];
        s3 += xv * weff[k * HEADS + 3];
    }
    a[n * HEADS + 0] = s0;
    a[n * HEADS + 1] = s1;
    a[n * HEADS + 2] = s2;
    a[n * HEADS + 3] = s3;
}

// ---------------------------------------------------------------------------
// 4) edge pass 1: segment max of leaky_relu logits per (dst, head)
// ---------------------------------------------------------------------------
__global__ void gat_edge_max(const int* __restrict__ src, const int* __restrict__ dst,
                             const float* __restrict__ a_s, const float* __restrict__ a_d,
                             float* __restrict__ amax, int E) {
    int e = blockIdx.x * blockDim.x + threadIdx.x;
    if (e >= E) return;
    int s = src[e], d = dst[e];
#pragma unroll
    for (int h = 0; h < HEADS; ++h) {
        float lg = leaky(a_s[s * HEADS + h] + a_d[d * HEADS + h]);
        atomicMaxF32(&amax[d * HEADS + h], lg);
    }
}

// ---------------------------------------------------------------------------
// 5) edge pass 2: ex = exp(logit - amax); store per-edge; denom += ex
// ---------------------------------------------------------------------------
__global__ void gat_edge_exp(const int* __restrict__ src, const int* __restrict__ dst,
                             const float* __restrict__ a_s, const float* __restrict__ a_d,
                             const float* __restrict__ amax, float* __restrict__ denom,
                             float* __restrict__ alphaE, int E) {
    int e = blockIdx.x * blockDim.x + threadIdx.x;
    if (e >= E) return;
    int s = src[e], d = dst[e];
#pragma unroll
    for (int h = 0; h < HEADS; ++h) {
        float lg = leaky(a_s[s * HEADS + h] + a_d[d * HEADS + h]);
        float ex = __expf(lg - amax[d * HEADS + h]);
        alphaE[(size_t)e * HEADS + h] = ex;
        atomicAdd(&denom[d * HEADS + h], ex);
    }
}

// ---------------------------------------------------------------------------
// 6) edge pass 3: wave-per-edge gather of hs[src] row (L2-resident), heads
//    reduced in registers, one atomicAdd per (edge, channel) into out.
// ---------------------------------------------------------------------------
__global__ __launch_bounds__(256) void gat_edge_agg(const int* __restrict__ src,
                                                    const int* __restrict__ dst,
                                                    const float* __restrict__ hs,
                                                    const float* __restrict__ alphaE,
                                                    const float* __restrict__ denom,
                                                    float* __restrict__ out, int E) {
    int e    = blockIdx.x * 8 + (threadIdx.x >> 5);
    int lane = threadIdx.x & 31;
    if (e >= E) return;                       // wave-uniform
    int s = src[e], d = dst[e];
    float w[HEADS];
#pragma unroll
    for (int h = 0; h < HEADS; ++h)
        w[h] = alphaE[(size_t)e * HEADS + h] /
               (denom[d * HEADS + h] + SM_EPS) * (1.0f / HEADS);
    const float* hrow = hs + (size_t)s * HC;
#pragma unroll
    for (int cc = 0; cc < 2; ++cc) {
        int c = lane + 32 * cc;
        float v = 0.f;
#pragma unroll
        for (int h = 0; h < HEADS; ++h)
            v += w[h] * hrow[h * OUT_DIM + c];
        atomicAdd(&out[(size_t)d * OUT_DIM + c], v);
    }
}

// ---------------------------------------------------------------------------
extern "C" void kernel_launch(void* const* d_in, const int* in_sizes, int n_in,
                              void* d_out, int out_size, void* d_ws, size_t ws_size,
                              hipStream_t stream) {
    (void)n_in; (void)out_size; (void)ws_size;
    const float* x_src   = (const float*)d_in[0];
    const float* x_dst   = (const float*)d_in[1];
    const int*   eidx    = (const int*)d_in[2];
    const float* W_src   = (const float*)d_in[3];
    const float* W_dst   = (const float*)d_in[4];
    const float* att_src = (const float*)d_in[5];
    const float* att_dst = (const float*)d_in[6];
    const float* bias    = (const float*)d_in[7];
    float* out = (float*)d_out;

    int nsrc = in_sizes[0] / IN_DIM;
    int ndst = in_sizes[1] / IN_DIM;
    int E    = in_sizes[2] / 2;
    const int* src = eidx;
    const int* dst = eidx + E;

    // workspace carve-up (floats)
    float* ws     = (float*)d_ws;
    float* hs     = ws;                            // nsrc*256
    float* a_s    = hs     + (size_t)nsrc * HC;    // nsrc*4
    float* a_d    = a_s    + (size_t)nsrc * HEADS; // ndst*4
    float* amax   = a_d    + (size_t)ndst * HEADS; // ndst*4
    float* denom  = amax   + (size_t)ndst * HEADS; // ndst*4
    float* alphaE = denom  + (size_t)ndst * HEADS; // E*4
    float* weffS  = alphaE + (size_t)E * HEADS;    // 512
    float* weffD  = weffS  + IN_DIM * HEADS;       // 512

    int initN = ndst * OUT_DIM;
    gat_init<<<(initN + 255) / 256, 256, 0, stream>>>(out, amax, denom, bias, ndst);

    gat_weff<<<1, IN_DIM, 0, stream>>>(W_src, att_src, weffS);
    gat_weff<<<1, IN_DIM, 0, stream>>>(W_dst, att_dst, weffD);

    int mtiles = (nsrc + 15) / 16;                 // 4 waves per block = one M-tile
    gat_gemm_hs<<<mtiles, 128, 0, stream>>>(x_src, W_src, hs, nsrc);

    gat_node_logits<<<(nsrc + 255) / 256, 256, 0, stream>>>(x_src, weffS, a_s, nsrc);
    gat_node_logits<<<(ndst + 255) / 256, 256, 0, stream>>>(x_dst, weffD, a_d, ndst);

    gat_edge_max<<<(E + 255) / 256, 256, 0, stream>>>(src, dst, a_s, a_d, amax, E);
    gat_edge_exp<<<(E + 255) / 256, 256, 0, stream>>>(src, dst, a_s, a_d, amax, denom, alphaE, E);
    gat_edge_agg<<<(E + 7) / 8, 256, 0, stream>>>(src, dst, hs, alphaE, denom, out, E);
}